// BoundarySeg_65652870086717
// MI455X (gfx1250) — compile-verified
//
#include <hip/hip_runtime.h>
#include <hip/hip_bf16.h>

typedef __attribute__((ext_vector_type(2))) float v2f;
typedef __attribute__((ext_vector_type(8))) float v8f;

namespace {
constexpr int Lc    = 1024;   // sequence length
constexpr int Hc    = 768;    // hidden
constexpr int SPAN  = 6;      // MAX_SPAN_LEN
constexpr int JT    = 16;     // j-rows per tile (WMMA M)
constexpr int KPAD  = 24;     // band K extent padded to 6 chunks of 4
constexpr int WAVES = 8;      // 256 threads
constexpr int HT_PER_WAVE = (Hc / 16) / WAVES;  // 48 h-tiles / 8 waves = 6
}

// select w[d] for dynamic d in [0,6), else 0 — pure VALU cndmask chain
__device__ __forceinline__ float band_pick(const float w[SPAN], int d) {
    float r = 0.0f;
    r = (d == 0) ? w[0] : r;
    r = (d == 1) ? w[1] : r;
    r = (d == 2) ? w[2] : r;
    r = (d == 3) ? w[3] : r;
    r = (d == 4) ? w[4] : r;
    r = (d == 5) ? w[5] : r;
    return r;
}

__global__ __launch_bounds__(256)
void BoundarySeg_banded_wmma_kernel(const float* __restrict__ span,
                                    const float* __restrict__ bh,
                                    float* __restrict__ out) {
    const int tile = blockIdx.x;          // B * L/16 tiles
    const int b    = tile >> 6;           // tile / (L/16)
    const int j0   = (tile & 63) << 4;    // 16-row j tile origin

    const int lane = threadIdx.x & 31;
    const int wave = threadIdx.x >> 5;
    const int m    = lane & 15;           // A-matrix row / B,C column within half-wave
    const int half = lane >> 4;           // 0: K/M low pair, 1: K/M high pair

    // ---- band weights for row j0+m (duplicated across half-waves) ----
    const float* spanRow = span + ((size_t)b * Lc + (size_t)(j0 + m)) * Lc;
    float w[SPAN];
#pragma unroll
    for (int d = 0; d < SPAN; ++d) {
        int jj  = j0 + m + d;
        int idx = (jj < Lc) ? jj : (Lc - 1);       // min(j+d, L-1)
        float v = spanRow[idx];
        w[d] = (jj < Lc) ? v : 0.0f;               // validity mask folded into w
    }
    const float wsum = ((w[0] + w[1]) + (w[2] + w[3])) + (w[4] + w[5]);

    // ---- broadcast wsum for the 8 C-rows this lane will store (loop-invariant:
    //      hoisted so no ds_bpermute appears inside the h-tile loop) ----
    float ws[8];
#pragma unroll
    for (int v = 0; v < 8; ++v) {
        ws[v] = __shfl(wsum, v + half * 8, 32);
    }

    // ---- A fragments: A[m,k] = w[k-m] if 0<=k-m<6 else 0, K padded to 24 ----
    // f32 16x4 A layout: lanes 0-15 hold K={0,1}, lanes 16-31 hold K={2,3}
    v2f afrag[KPAD / 4];
#pragma unroll
    for (int kc = 0; kc < KPAD / 4; ++kc) {
        const int k0 = kc * 4 + half * 2;
        afrag[kc].x = band_pick(w, k0 - m);
        afrag[kc].y = band_pick(w, k0 + 1 - m);
    }

    const float* bhB  = bh  + (size_t)b * Lc * Hc;
    float*       outB = out + (size_t)b * Lc * (2 * Hc);

    // ---- sweep this wave's h-tiles ----
#pragma unroll 1
    for (int t = 0; t < HT_PER_WAVE; ++t) {
        const int ht  = wave * HT_PER_WAVE + t;
        const int col = ht * 16 + m;

        v8f acc = {};
#pragma unroll
        for (int kc = 0; kc < KPAD / 4; ++kc) {
            const int k0 = kc * 4 + half * 2;
            int r0 = j0 + k0;     r0 = (r0 < Lc) ? r0 : (Lc - 1);  // clamp; masked by A
            int r1 = j0 + k0 + 1; r1 = (r1 < Lc) ? r1 : (Lc - 1);
            v2f bfrag;
            bfrag.x = bhB[(size_t)r0 * Hc + col];   // B VGPR0: K = k0
            bfrag.y = bhB[(size_t)r1 * Hc + col];   // B VGPR1: K = k0+1
            acc = __builtin_amdgcn_wmma_f32_16x16x4_f32(
                /*neg_a=*/false, afrag[kc],
                /*neg_b=*/false, bfrag,
                /*c_mod=*/(short)0, acc,
                /*reuse_a=*/false, /*reuse_b=*/false);
        }

        // first half of output: C layout VGPR v -> M = v + half*8, N = col
#pragma unroll
        for (int v = 0; v < 8; ++v) {
            const int row = v + half * 8;
            outB[(size_t)(j0 + row) * (2 * Hc) + col] = acc[v];
        }

        // second half: BH[j,h] * wsum[j] with pre-broadcast wsum
#pragma unroll
        for (int v = 0; v < 8; ++v) {
            const int row = v + half * 8;
            const float val = bhB[(size_t)(j0 + row) * Hc + col] * ws[v];
            outB[(size_t)(j0 + row) * (2 * Hc) + Hc + col] = val;
        }
    }
}

extern "C" void kernel_launch(void* const* d_in, const int* in_sizes, int n_in,
                              void* d_out, int out_size, void* d_ws, size_t ws_size,
                              hipStream_t stream) {
    (void)in_sizes; (void)n_in; (void)d_ws; (void)ws_size; (void)out_size;
    const float* span = (const float*)d_in[0];   // (16,1024,1024,1) f32
    const float* bh   = (const float*)d_in[1];   // (16,1024,768)    f32
    float*       outp = (float*)d_out;           // (16,1024,1536)   f32

    const int B = 16;
    const dim3 grid(B * (Lc / JT));              // 1024 tiles
    const dim3 block(32 * WAVES);                // 256 threads, 8 wave32
    BoundarySeg_banded_wmma_kernel<<<grid, block, 0, stream>>>(span, bh, outp);
}